// PyCoxLoss_29386166239328
// MI455X (gfx1250) — compile-verified
//
#include <hip/hip_runtime.h>
#include <stdint.h>

// ---------------------------------------------------------------------------
// Cox partial-likelihood loss, N = 8192, single fused workgroup kernel.
// LDS-resident: TDM DMA loads -> bitonic KV sort -> stabilized reverse
// cum-logsumexp scan -> reduction -> scalar out.
// ---------------------------------------------------------------------------

typedef __attribute__((ext_vector_type(4))) unsigned int u32x4;
typedef __attribute__((ext_vector_type(4))) int          i32x4;
typedef __attribute__((ext_vector_type(8))) int          i32x8;

#define NTOT   8192
#define NTHR   1024
#define PERTHR (NTOT / NTHR)

// LDS layout (dynamic shared, 128 KB total):
//   [0      , 65536 ) u64 keys[8192]          (sort area)
//   [65536  , 98304 ) float staging pred[8192]
//   [98304  , 131072) float staging ytime[8192]
// After packing, staging region is reused:
//   [65536  , 69632 ) float redA[1024]
//   [69632  , 73728 ) float redB[1024]
#define LDS_KEYS   0u
#define LDS_PRED   65536u
#define LDS_YTIME  98304u
#define LDS_BYTES  131072u

#if __has_builtin(__builtin_amdgcn_tensor_load_to_lds) && \
    __has_builtin(__builtin_amdgcn_s_wait_tensorcnt)
#define USE_TDM 1
#else
#define USE_TDM 0
#endif

#if USE_TDM
// Issue one TDM descriptor: 1-D tile of n fp32 elements, global -> LDS.
__device__ __forceinline__ void tdm_load_1d_f32(const void* gptr,
                                                unsigned lds_byte_off,
                                                unsigned n) {
  uint64_t ga = (uint64_t)gptr;

  // D# group 0 (128b): count=1 | lds_addr | global_addr[56:0] | type=2
  u32x4 g0;
  g0.x = 1u;                                   // count=1, user-mode, no gather
  g0.y = lds_byte_off;                         // lds_addr (bytes)
  g0.z = (unsigned)(ga & 0xFFFFFFFFu);         // global_addr[31:0]
  g0.w = (unsigned)((ga >> 32) & 0x01FFFFFFu)  // global_addr[56:32]
       | (2u << 30);                           // type = 2 ("image")

  // D# group 1 (256b)
  i32x8 g1;
  g1[0] = (int)(2u << 16);                     // data_size = 2 (4 bytes)
  g1[1] = (int)((n & 0xFFFFu) << 16);          // tensor_dim0[15:0]
  g1[2] = (int)((n >> 16) | (1u << 16));       // tensor_dim0[31:16] | tensor_dim1 = 1
  g1[3] = (int)((n & 0xFFFFu) << 16);          // tile_dim0 = n
  g1[4] = 1;                                   // tile_dim1 = 1, tile_dim2 = 0
  g1[5] = (int)n;                              // tensor_dim0_stride[31:0] = n
  g1[6] = 0;                                   // stride hi / dim1_stride lo
  g1[7] = 0;

  i32x4 z4 = {0, 0, 0, 0};
#if defined(__clang_major__) && (__clang_major__ >= 23)
  i32x8 z8 = {0, 0, 0, 0, 0, 0, 0, 0};
  __builtin_amdgcn_tensor_load_to_lds(g0, g1, z4, z4, z8, 0);
#else
  __builtin_amdgcn_tensor_load_to_lds(g0, g1, z4, z4, 0);
#endif
}
#endif // USE_TDM

__global__ __launch_bounds__(NTHR)
void PyCoxLoss_29386166239328_kernel(const float* __restrict__ pred,
                                     const float* __restrict__ ytime,
                                     float* __restrict__ out, int n) {
  extern __shared__ unsigned char smem[];
  unsigned long long* keys = (unsigned long long*)(smem + LDS_KEYS);
  float* spred = (float*)(smem + LDS_PRED);
  float* syt   = (float*)(smem + LDS_YTIME);
  float* redA  = (float*)(smem + LDS_PRED);          // reuse after packing
  float* redB  = (float*)(smem + LDS_PRED + 4096u);  // reuse after packing

  const unsigned tid = threadIdx.x;

  // ---- Stage inputs into LDS --------------------------------------------
#if USE_TDM
  if (tid < 32u) {  // wave 0 only: TDM ignores EXEC, avoid 32 duplicate DMAs
    tdm_load_1d_f32(pred,  LDS_PRED,  (unsigned)n);
    tdm_load_1d_f32(ytime, LDS_YTIME, (unsigned)n);
    __builtin_amdgcn_s_wait_tensorcnt(0);
  }
#else
  for (unsigned i = tid; i < (unsigned)n; i += NTHR) {
    spred[i] = pred[i];
    syt[i]   = ytime[i];
  }
#endif
  __syncthreads();

  // ---- Pack (ytime_bits << 32) | pred_bits; ytime > 0 => bits monotonic --
  for (unsigned k = 0; k < PERTHR; ++k) {
    unsigned i  = tid + k * NTHR;
    unsigned kb = __float_as_uint(syt[i]);
    unsigned vb = __float_as_uint(spred[i]);
    keys[i] = ((unsigned long long)kb << 32) | (unsigned long long)vb;
  }
  __syncthreads();

  // ---- Bitonic sort ascending over 8192 u64 keys ------------------------
  for (unsigned kk = 2; kk <= NTOT; kk <<= 1) {
    for (unsigned j = kk >> 1; j > 0; j >>= 1) {
      for (unsigned c = tid; c < (NTOT / 2); c += NTHR) {
        unsigned i = (c << 1) - (c & (j - 1));
        unsigned p = i + j;
        bool asc = ((i & kk) == 0);
        unsigned long long a = keys[i], b = keys[p];
        if ((a > b) == asc) { keys[i] = b; keys[p] = a; }
      }
      __syncthreads();
    }
  }

  // ---- Load this thread's chunk in REVERSED order; max & sum reductions --
  // g-index gi scans the reversed array: sp_rev[gi] = sp_sorted[N-1-gi].
  float spv[PERTHR];
  float lmax = -3.402823466e38f, lsum = 0.f;
  for (unsigned k = 0; k < PERTHR; ++k) {
    unsigned gi = tid * PERTHR + k;
    unsigned i  = (NTOT - 1u) - gi;
    float v = __uint_as_float((unsigned)(keys[i] & 0xFFFFFFFFull));
    spv[k] = v;
    lmax = fmaxf(lmax, v);
    lsum += v;
  }
  redA[tid] = lmax;
  redB[tid] = lsum;
  __syncthreads();
  for (unsigned s = NTHR >> 1; s > 0; s >>= 1) {
    if (tid < s) {
      redA[tid] = fmaxf(redA[tid], redA[tid + s]);
      redB[tid] += redB[tid + s];
    }
    __syncthreads();
  }
  const float M     = redA[0];
  const float sumSp = redB[0];
  __syncthreads();

  // ---- Forward scan of exp(sp_rev - M): S_j = revcumsum of exp ----------
  float vals[PERTHR];
  float run = 0.f;
  for (unsigned k = 0; k < PERTHR; ++k) {
    run += expf(spv[k] - M);
    vals[k] = run;  // local inclusive prefix
  }
  redA[tid] = run;  // chunk totals
  __syncthreads();

  // Hillis-Steele inclusive scan of 1024 chunk totals in LDS.
  const float own = run;
  for (unsigned off = 1; off < NTHR; off <<= 1) {
    float y  = redA[tid];
    float yo = (tid >= off) ? redA[tid - off] : 0.f;
    __syncthreads();
    redA[tid] = y + yo;
    __syncthreads();
  }
  const float excl = redA[tid] - own;

  // ---- Sum of log(S_j) over this chunk, then reduce ---------------------
  float llog = 0.f;
  for (unsigned k = 0; k < PERTHR; ++k) {
    llog += logf(vals[k] + excl);
  }
  redB[tid] = llog;
  __syncthreads();
  for (unsigned s = NTHR >> 1; s > 0; s >>= 1) {
    if (tid < s) redB[tid] += redB[tid + s];
    __syncthreads();
  }

  if (tid == 0) {
    float fn = (float)n;
    out[0] = (redB[0] + fn * M - sumSp) / fn;
  }
}

extern "C" void kernel_launch(void* const* d_in, const int* in_sizes, int n_in,
                              void* d_out, int out_size, void* d_ws, size_t ws_size,
                              hipStream_t stream) {
  (void)n_in; (void)out_size; (void)d_ws; (void)ws_size;
  const float* pred  = (const float*)d_in[0];
  const float* ytime = (const float*)d_in[1];
  float* out = (float*)d_out;
  int n = in_sizes[0];  // 8192

  PyCoxLoss_29386166239328_kernel<<<dim3(1), dim3(NTHR), LDS_BYTES, stream>>>(
      pred, ytime, out, n);
}